// GCN_68118181314631
// MI455X (gfx1250) — compile-verified
//
#include <hip/hip_runtime.h>
#include <hip/hip_bf16.h>

// ---------------------------------------------------------------------------
// GCN (2-layer) for MI455X / gfx1250.
//   deg -> dinv -> norm              (memory-light elementwise/atomic)
//   h  = x @ W1  via V_WMMA_F32_16X16X4_F32  (fp32 tensor op, full precision)
//   agg1 = scatter_add(h[src]*norm)  (wave-per-edge, float4 + atomic f32)
//   agg1 = relu(agg1 + b1)
//   h  = agg1 @ W2 (WMMA again, h buffer reused)
//   out = scatter_add(h[src]*norm) + b2
// Edge phase dominates (~1.8 GB traffic -> ~80us @ 23.3 TB/s); GEMMs are
// ~3.3 GFLOP and fully L2-resident (W = 64KB, node matrix = 25.6MB << 192MB).
// ---------------------------------------------------------------------------

#define N_NODES 50000
#define N_EDGES 800000
#define DIM     128

typedef float v2f __attribute__((ext_vector_type(2)));
typedef float v8f __attribute__((ext_vector_type(8)));

// ---------------- elementwise / setup kernels ------------------------------

__global__ void k_zero_f32(float* __restrict__ p, int n) {
    int stride = gridDim.x * blockDim.x;
    for (int i = blockIdx.x * blockDim.x + threadIdx.x; i < n; i += stride)
        p[i] = 0.0f;
}

__global__ void k_deg(const int* __restrict__ dst, const float* __restrict__ ew,
                      float* __restrict__ deg, int nE) {
    int e = blockIdx.x * blockDim.x + threadIdx.x;
    if (e < nE) atomicAdd(&deg[dst[e]], ew[e]);
}

__global__ void k_dinv(const float* __restrict__ deg, float* __restrict__ dinv, int n) {
    int i = blockIdx.x * blockDim.x + threadIdx.x;
    if (i < n) {
        float d = deg[i];
        dinv[i] = (d > 0.0f) ? __frsqrt_rn(fmaxf(d, 1e-30f)) : 0.0f;
    }
}

__global__ void k_norm(const int* __restrict__ src, const int* __restrict__ dst,
                       const float* __restrict__ ew, const float* __restrict__ dinv,
                       float* __restrict__ norm, int nE) {
    int e = blockIdx.x * blockDim.x + threadIdx.x;
    if (e < nE) norm[e] = dinv[src[e]] * ew[e] * dinv[dst[e]];
}

// ---------------- fp32 WMMA GEMM: H[M x 128] = A[M x 128] * W[128 x 128] ---
// Block = 256 threads = 8 waves; block handles 16 rows, each wave one 16x16
// tile (wave w -> cols [16w,16w+16)). K looped in steps of 4 with
// V_WMMA_F32_16X16X4_F32. A-slab (16x128 = 8KB) staged in LDS, shared by all
// 8 waves; B fragments read from L2-resident W.
__global__ __launch_bounds__(256) void k_gemm_wmma(const float* __restrict__ A,
                                                   const float* __restrict__ W,
                                                   float* __restrict__ H) {
    __shared__ float As[16 * DIM];                 // 8 KB of 320 KB/WGP
    const int r0 = blockIdx.x * 16;

    // cooperative slab load: 2048 floats = 512 float4 = 2 per thread
    {
        const float4* s = (const float4*)(A + (size_t)r0 * DIM);
        float4* d = (float4*)As;
        d[threadIdx.x]       = s[threadIdx.x];
        d[threadIdx.x + 256] = s[threadIdx.x + 256];
    }
    __syncthreads();

    const int lane = threadIdx.x & 31;             // wave32
    const int wv   = threadIdx.x >> 5;
    const int c0   = wv * 16;
    const int half = lane >> 4;                    // 0: K lo pair, 1: K hi pair
    const int l16  = lane & 15;

    v8f acc = {};
#pragma unroll
    for (int k0 = 0; k0 < DIM; k0 += 4) {
        const int ka = k0 + half * 2;
        // A 16x4 f32 frag: lanes 0-15 -> M=l16,K={k0,k0+1}; lanes 16-31 -> K={k0+2,k0+3}
        v2f a;
        a.x = As[l16 * DIM + ka];
        a.y = As[l16 * DIM + ka + 1];
        // B 4x16 f32 frag: row K striped across lanes, same half split as A
        v2f b;
        b.x = W[(size_t)ka * DIM + c0 + l16];
        b.y = W[(size_t)(ka + 1) * DIM + c0 + l16];
        acc = __builtin_amdgcn_wmma_f32_16x16x4_f32(
            /*neg_a=*/false, a, /*neg_b=*/false, b,
            /*c_mod=*/(short)0, acc, /*reuse_a=*/false, /*reuse_b=*/false);
    }

    // C/D layout: VGPR i -> M = i (lanes 0-15) or i+8 (lanes 16-31), N = l16
    float* out = H + (size_t)r0 * DIM + c0;
#pragma unroll
    for (int i = 0; i < 8; ++i)
        out[(size_t)(i + half * 8) * DIM + l16] = acc[i];
}

// ---------------- edge gather * norm -> atomic scatter-add -----------------
// One wave per edge: 32 lanes x float4 covers the 128-wide row. Coalesced
// 512B gather, 4 global_atomic_add_f32 per lane.
__global__ __launch_bounds__(256) void k_edge_scatter(const float* __restrict__ h,
                                                      const int* __restrict__ src,
                                                      const int* __restrict__ dst,
                                                      const float* __restrict__ norm,
                                                      float* __restrict__ agg, int nE) {
    int e = blockIdx.x * (blockDim.x >> 5) + (threadIdx.x >> 5);
    if (e >= nE) return;
    const int lane = threadIdx.x & 31;
    const int s = src[e];
    const int d = dst[e];
    const float w = norm[e];
    float4 v = ((const float4*)(h + (size_t)s * DIM))[lane];
    float* a = agg + (size_t)d * DIM + lane * 4;
    atomicAdd(a + 0, v.x * w);
    atomicAdd(a + 1, v.y * w);
    atomicAdd(a + 2, v.z * w);
    atomicAdd(a + 3, v.w * w);
}

__global__ void k_bias_relu(float* __restrict__ p, const float* __restrict__ b, int n) {
    int i = blockIdx.x * blockDim.x + threadIdx.x;
    if (i < n) p[i] = fmaxf(p[i] + b[i & (DIM - 1)], 0.0f);
}

__global__ void k_bias(float* __restrict__ p, const float* __restrict__ b, int n) {
    int i = blockIdx.x * blockDim.x + threadIdx.x;
    if (i < n) p[i] = p[i] + b[i & (DIM - 1)];
}

// ---------------------------------------------------------------------------

extern "C" void kernel_launch(void* const* d_in, const int* in_sizes, int n_in,
                              void* d_out, int out_size, void* d_ws, size_t ws_size,
                              hipStream_t stream) {
    const float* x  = (const float*)d_in[0];
    const int*   ei = (const int*)d_in[1];          // [2, E] flat
    const float* ew = (const float*)d_in[2];
    const float* W1 = (const float*)d_in[3];
    const float* b1 = (const float*)d_in[4];
    const float* W2 = (const float*)d_in[5];
    const float* b2 = (const float*)d_in[6];
    const int* src = ei;
    const int* dst = ei + N_EDGES;
    float* out = (float*)d_out;

    // workspace layout (floats), 16B-aligned chunks
    float* ws   = (float*)d_ws;
    float* deg  = ws;                     // 50048
    float* dinv = deg  + 50048;           // 50048
    float* norm = dinv + 50048;           // 800000
    float* h    = norm + 800000;          // 6,400,000 (h1, reused as h2)
    float* agg1 = h    + (size_t)N_NODES * DIM;  // 6,400,000
    const int NF = N_NODES * DIM;         // 6,400,000

    // ---- zero accumulators (ws/d_out arrive poisoned) ----
    k_zero_f32<<<2048, 256, 0, stream>>>(deg, N_NODES);
    k_zero_f32<<<2048, 256, 0, stream>>>(agg1, NF);
    k_zero_f32<<<2048, 256, 0, stream>>>(out, NF);

    // ---- normalization coefficients ----
    k_deg<<<(N_EDGES + 255) / 256, 256, 0, stream>>>(dst, ew, deg, N_EDGES);
    k_dinv<<<(N_NODES + 255) / 256, 256, 0, stream>>>(deg, dinv, N_NODES);
    k_norm<<<(N_EDGES + 255) / 256, 256, 0, stream>>>(src, dst, ew, dinv, norm, N_EDGES);

    // ---- layer 1: projection (WMMA) -> edge scatter -> bias+relu ----
    k_gemm_wmma<<<N_NODES / 16, 256, 0, stream>>>(x, W1, h);
    k_edge_scatter<<<N_EDGES / 8, 256, 0, stream>>>(h, src, dst, norm, agg1, N_EDGES);
    k_bias_relu<<<(NF + 255) / 256, 256, 0, stream>>>(agg1, b1, NF);

    // ---- layer 2: projection (WMMA, h reused) -> edge scatter -> bias ----
    k_gemm_wmma<<<N_NODES / 16, 256, 0, stream>>>(agg1, W2, h);
    k_edge_scatter<<<N_EDGES / 8, 256, 0, stream>>>(h, src, dst, norm, out, N_EDGES);
    k_bias<<<(NF + 255) / 256, 256, 0, stream>>>(out, b2, NF);
}